// PfoTGNRec_82944408420705
// MI455X (gfx1250) — compile-verified
//
#include <hip/hip_runtime.h>
#include <math.h>

typedef __attribute__((ext_vector_type(2))) float v2f;
typedef __attribute__((ext_vector_type(8))) float v8f;

#define B_SZ 4096
#define D_SZ 64
#define NI_SZ 10000
#define K_IH 130
#define N3D 192

__device__ __forceinline__ v8f wmma_f32(v2f a, v2f b, v8f c) {
  // D = A(16x4) * B(4x16) + C(16x16), all f32.
  // (neg_a, A, neg_b, B, c_mod, C, reuse_a, reuse_b)
  return __builtin_amdgcn_wmma_f32_16x16x4_f32(false, a, false, b, (short)0, c,
                                               false, false);
}

__device__ __forceinline__ v2f ldv2(const float* p) {
  return *(const v2f*)p;  // 8-byte aligned by construction -> global_load_b64
}

// ---------------- GRU GEMMs: gi = msg @ W_ih^T + b_ih ; gh = h @ W_hh^T + b_hh
// 8 waves/block, one 16x16 C-tile per wave. tiles = 256*12 = 3072 = 384 blocks.
__global__ __launch_bounds__(256) void k_gemm_gru(
    const int* __restrict__ u, const int* __restrict__ v,
    const float* __restrict__ delta_t, const float* __restrict__ edge,
    const float* __restrict__ uemb, const float* __restrict__ iemb,
    const float* __restrict__ mem,
    const float* __restrict__ w_ih, const float* __restrict__ w_hh,
    const float* __restrict__ b_ih, const float* __restrict__ b_hh,
    float* __restrict__ gi_out, float* __restrict__ gh_out) {
  const int MT = B_SZ / 16;  // 256 m-tiles
  int lane = threadIdx.x & 31;
  int tile = blockIdx.x * 8 + (threadIdx.x >> 5);
  int m0 = (tile % MT) * 16;
  int n0 = (tile / MT) * 16;

  int am = m0 + (lane & 15);
  int k_half = (lane >> 4) << 1;  // lanes 0-15: K {k0,k0+1}; 16-31: {k0+2,k0+3}
  bool lo16 = (lane < 16);

  const float* urow = uemb + (long)u[am] * D_SZ;
  const float* irow = iemb + (long)v[am] * D_SZ;
  const float* hrow = mem  + (long)u[am] * D_SZ;
  float dt = delta_t[am];
  float ef = edge[am];  // E == 1

  int bn = n0 + (lane & 15);
  const float* wih_row = w_ih + (long)bn * K_IH;
  const float* whh_row = w_hh + (long)bn * D_SZ;

  v8f acc_i = {};
  v8f acc_h = {};

  // gi phase 1: msg k = 0..63 (u_emb)
  for (int k0 = 0; k0 < 64; k0 += 4) {
    int ka = k0 + k_half;
    acc_i = wmma_f32(ldv2(urow + ka), ldv2(wih_row + ka), acc_i);
  }
  // gi phase 2: msg k = 64..127 (v_emb)
  for (int k0 = 0; k0 < 64; k0 += 4) {
    int ka = k0 + k_half;
    acc_i = wmma_f32(ldv2(irow + ka), ldv2(wih_row + 64 + ka), acc_i);
  }
  // gi tail: k = 128 (delta_t), 129 (edge_feat), 130..131 zero-pad
  {
    v2f a, b;
    a.x = lo16 ? dt : 0.f;
    a.y = lo16 ? ef : 0.f;
    b.x = lo16 ? wih_row[128] : 0.f;
    b.y = lo16 ? wih_row[129] : 0.f;
    acc_i = wmma_f32(a, b, acc_i);
  }

  // gh: K = 64
  for (int k0 = 0; k0 < 64; k0 += 4) {
    int ka = k0 + k_half;
    acc_h = wmma_f32(ldv2(hrow + ka), ldv2(whh_row + ka), acc_h);
  }

  // C/D layout: VGPR r -> row m0+r (lanes 0-15) / m0+8+r (lanes 16-31)
  int cm = m0 + ((lane >> 4) << 3);
  int cn = n0 + (lane & 15);
  float bi = b_ih[cn], bh = b_hh[cn];
#pragma unroll
  for (int r = 0; r < 8; ++r) {
    gi_out[(long)(cm + r) * N3D + cn] = acc_i[r] + bi;
    gh_out[(long)(cm + r) * N3D + cn] = acc_h[r] + bh;
  }
}

// ---------------- GRU elementwise -> h_new
__global__ __launch_bounds__(256) void k_gru_elem(
    const float* __restrict__ gi, const float* __restrict__ gh,
    const int* __restrict__ u, const float* __restrict__ mem,
    float* __restrict__ h_new) {
  int tid = blockIdx.x * blockDim.x + threadIdx.x;
  if (tid >= B_SZ * D_SZ) return;
  int m = tid >> 6, d = tid & 63;
  const float* gir = gi + (long)m * N3D;
  const float* ghr = gh + (long)m * N3D;
  float i_r = gir[d], i_z = gir[64 + d], i_n = gir[128 + d];
  float h_r = ghr[d], h_z = ghr[64 + d], h_n = ghr[128 + d];
  float r = 1.f / (1.f + __expf(-(i_r + h_r)));
  float z = 1.f / (1.f + __expf(-(i_z + h_z)));
  float n = tanhf(i_n + r * h_n);
  float h = mem[(long)u[m] * D_SZ + d];
  h_new[tid] = (1.f - z) * n + z * h;
}

// ---------------- qkv = x @ W_in^T + b_in ; x rows: 2m -> u_emb, 2m+1 -> v_emb+edge
// tiles = 512*12 = 6144 = 768 blocks of 8 waves.
__global__ __launch_bounds__(256) void k_qkv(
    const int* __restrict__ u, const int* __restrict__ v,
    const float* __restrict__ edge,
    const float* __restrict__ uemb, const float* __restrict__ iemb,
    const float* __restrict__ w_in, const float* __restrict__ b_in,
    float* __restrict__ qkv) {
  const int MT = 2 * B_SZ / 16;  // 512 row-tiles
  int lane = threadIdx.x & 31;
  int tile = blockIdx.x * 8 + (threadIdx.x >> 5);
  int r0 = (tile % MT) * 16;
  int n0 = (tile / MT) * 16;

  int ar = r0 + (lane & 15);
  int m = ar >> 1, s = ar & 1;
  const float* src = s ? (iemb + (long)v[m] * D_SZ) : (uemb + (long)u[m] * D_SZ);
  float add = s ? edge[m] : 0.f;
  int k_half = (lane >> 4) << 1;
  int bn = n0 + (lane & 15);
  const float* wrow = w_in + (long)bn * D_SZ;

  v8f acc = {};
  for (int k0 = 0; k0 < 64; k0 += 4) {
    int ka = k0 + k_half;
    v2f sa = ldv2(src + ka);
    v2f a;
    a.x = sa.x + add;
    a.y = sa.y + add;
    acc = wmma_f32(a, ldv2(wrow + ka), acc);
  }
  int cm = r0 + ((lane >> 4) << 3);
  int cn = n0 + (lane & 15);
  float bb = b_in[cn];
#pragma unroll
  for (int r = 0; r < 8; ++r)
    qkv[(long)(cm + r) * N3D + cn] = acc[r] + bb;
}

// ---------------- per-(b, head) 2-key attention; only q-position 0 needed
__global__ __launch_bounds__(64) void k_attn(const float* __restrict__ qkv,
                                             float* __restrict__ o0) {
  int b = blockIdx.x;
  int t = threadIdx.x;
  int h = t >> 4, dd = t & 15;
  const float* r0 = qkv + (long)(2 * b) * N3D;
  const float* r1 = qkv + (long)(2 * b + 1) * N3D;
  float s0 = 0.f, s1 = 0.f;
  int o = h * 16;
#pragma unroll
  for (int j = 0; j < 16; ++j) {
    float q = r0[o + j];
    s0 += q * r0[64 + o + j];
    s1 += q * r1[64 + o + j];
  }
  s0 *= 0.25f; s1 *= 0.25f;
  float mx = fmaxf(s0, s1);
  float e0 = __expf(s0 - mx), e1 = __expf(s1 - mx);
  float inv = 1.f / (e0 + e1);
  o0[(long)b * D_SZ + t] = (e0 * r0[128 + o + dd] + e1 * r1[128 + o + dd]) * inv;
}

// ---------------- z_u = o0 @ W_out^T + b_out ; tiles = 256*4 = 1024 = 128 blocks
__global__ __launch_bounds__(256) void k_outproj(
    const float* __restrict__ o0, const float* __restrict__ w,
    const float* __restrict__ bias, float* __restrict__ z_u) {
  const int MT = B_SZ / 16;  // 256 m-tiles
  int lane = threadIdx.x & 31;
  int tile = blockIdx.x * 8 + (threadIdx.x >> 5);
  int m0 = (tile % MT) * 16;
  int n0 = (tile / MT) * 16;

  int am = m0 + (lane & 15);
  int k_half = (lane >> 4) << 1;
  int bn = n0 + (lane & 15);
  const float* arow = o0 + (long)am * D_SZ;
  const float* wrow = w + (long)bn * D_SZ;
  v8f acc = {};
  for (int k0 = 0; k0 < 64; k0 += 4) {
    int ka = k0 + k_half;
    acc = wmma_f32(ldv2(arow + ka), ldv2(wrow + ka), acc);
  }
  int cm = m0 + ((lane >> 4) << 3);
  int cn = n0 + (lane & 15);
  float bb = bias[cn];
#pragma unroll
  for (int r = 0; r < 8; ++r)
    z_u[(long)(cm + r) * D_SZ + cn] = acc[r] + bb;
}

// ---------------- exact JAX threefry2x32 uniform, key(42), n=10000
__device__ __forceinline__ unsigned rotl32(unsigned x, int d) {
  return (x << d) | (x >> (32 - d));
}
__device__ float jax_uniform01_key42(int i) {
  const int half = NI_SZ / 2;
  bool second = (i >= half);
  int j = second ? (i - half) : i;
  unsigned x0 = (unsigned)j, x1 = (unsigned)(j + half);
  unsigned ks0 = 0u, ks1 = 42u, ks2 = 0x1BD11BDAu ^ 0u ^ 42u;
  x0 += ks0; x1 += ks1;
#define TF_ROUND(r) { x0 += x1; x1 = rotl32(x1, r); x1 ^= x0; }
  TF_ROUND(13) TF_ROUND(15) TF_ROUND(26) TF_ROUND(6)
  x0 += ks1; x1 += ks2 + 1u;
  TF_ROUND(17) TF_ROUND(29) TF_ROUND(16) TF_ROUND(24)
  x0 += ks2; x1 += ks0 + 2u;
  TF_ROUND(13) TF_ROUND(15) TF_ROUND(26) TF_ROUND(6)
  x0 += ks0; x1 += ks1 + 3u;
  TF_ROUND(17) TF_ROUND(29) TF_ROUND(16) TF_ROUND(24)
  x0 += ks1; x1 += ks2 + 4u;
  TF_ROUND(13) TF_ROUND(15) TF_ROUND(26) TF_ROUND(6)
  x0 += ks2; x1 += ks0 + 5u;
#undef TF_ROUND
  unsigned bits = second ? x1 : x0;
  union { unsigned u; float f; } cv;
  cv.u = (bits >> 9) | 0x3F800000u;
  return cv.f - 1.0f;
}

// ---------------- combined[i] = LAM*(returns/GAMMA - 0.5*mean_b cov[i,v[b]]) + (1-LAM)*pref
__global__ __launch_bounds__(256) void k_cov(
    const float* __restrict__ cov, const int* __restrict__ v,
    const float* __restrict__ returns, float* __restrict__ combined) {
  __shared__ int vsh[B_SZ];
  __shared__ float red[256];
  int i = blockIdx.x;
  int t = threadIdx.x;
  for (int b = t; b < B_SZ; b += 256) vsh[b] = v[b];
  __syncthreads();
  const float* row = cov + (long)i * NI_SZ;
  float s = 0.f;
  for (int b = t; b < B_SZ; b += 256) s += row[vsh[b]];
  red[t] = s;
  __syncthreads();
  for (int off = 128; off > 0; off >>= 1) {
    if (t < off) red[t] += red[t + off];
    __syncthreads();
  }
  if (t == 0) {
    float mean = red[0] * (1.0f / (float)B_SZ);
    float mv = returns[i] - 0.5f * mean;  // GAMMA = 1.0
    combined[i] = 0.5f * mv + 0.5f * jax_uniform01_key42(i);  // LAM = 0.5
  }
}

// ---------------- global top-5 of combined (scores row-invariant; mask is 1 idx/row)
__global__ __launch_bounds__(256) void k_top5(const float* __restrict__ combined,
                                              int* __restrict__ top_idx) {
  __shared__ float sval[256];
  __shared__ int sidx[256];
  __shared__ int found[5];
  int t = threadIdx.x;
  for (int pass = 0; pass < 5; ++pass) {
    float best = -__builtin_inff();
    int bidx = 0x7fffffff;
    for (int i = t; i < NI_SZ; i += 256) {
      bool skip = false;
      for (int p = 0; p < pass; ++p)
        if (found[p] == i) skip = true;
      if (skip) continue;
      float c = combined[i];
      if (c > best || (c == best && i < bidx)) { best = c; bidx = i; }
    }
    sval[t] = best; sidx[t] = bidx;
    __syncthreads();
    for (int off = 128; off > 0; off >>= 1) {
      if (t < off) {
        if (sval[t + off] > sval[t] ||
            (sval[t + off] == sval[t] && sidx[t + off] < sidx[t])) {
          sval[t] = sval[t + off]; sidx[t] = sidx[t + off];
        }
      }
      __syncthreads();
    }
    if (t == 0) { found[pass] = sidx[0]; top_idx[pass] = sidx[0]; }
    __syncthreads();
  }
}

// ---------------- per-row select top-4 excluding v[b]
__global__ __launch_bounds__(256) void k_select(const int* __restrict__ top_idx,
                                                const int* __restrict__ v,
                                                float* __restrict__ pos,
                                                float* __restrict__ neg) {
  int b = blockIdx.x * blockDim.x + threadIdx.x;
  if (b >= B_SZ) return;
  int vb = v[b];
  int cand[5] = {top_idx[0], top_idx[1], top_idx[2], top_idx[3], top_idx[4]};
  int sel[4];
  int c = 0;
#pragma unroll
  for (int p = 0; p < 5; ++p)
    if (cand[p] != vb && c < 4) sel[c++] = cand[p];
  pos[b] = (float)sel[0];
  neg[(long)b * 3 + 0] = (float)sel[1];
  neg[(long)b * 3 + 1] = (float)sel[2];
  neg[(long)b * 3 + 2] = (float)sel[3];
}

extern "C" void kernel_launch(void* const* d_in, const int* in_sizes, int n_in,
                              void* d_out, int out_size, void* d_ws,
                              size_t ws_size, hipStream_t stream) {
  const int*   u       = (const int*)d_in[0];
  const int*   v       = (const int*)d_in[1];
  // d_in[2] = t (unused by reference)
  const float* delta_t = (const float*)d_in[3];
  const float* edge    = (const float*)d_in[4];
  const float* returns = (const float*)d_in[5];
  const float* cov     = (const float*)d_in[6];
  const float* uemb    = (const float*)d_in[7];
  const float* iemb    = (const float*)d_in[8];
  const float* mem     = (const float*)d_in[9];
  const float* w_ih    = (const float*)d_in[10];
  const float* w_hh    = (const float*)d_in[11];
  const float* b_ih    = (const float*)d_in[12];
  const float* b_hh    = (const float*)d_in[13];
  const float* w_in    = (const float*)d_in[14];
  const float* b_in    = (const float*)d_in[15];
  const float* w_out   = (const float*)d_in[16];
  const float* b_out   = (const float*)d_in[17];

  float* out   = (float*)d_out;
  float* z_u   = out;                                // B*D
  float* h_new = out + (size_t)B_SZ * D_SZ;          // B*D
  float* pos   = out + (size_t)2 * B_SZ * D_SZ;      // B
  float* neg   = pos + B_SZ;                         // B*3

  float* ws   = (float*)d_ws;
  float* gi   = ws;                                  // B*192
  float* gh   = gi + (size_t)B_SZ * N3D;             // B*192
  float* qkv  = ws;                                  // 2B*192 (aliases gi+gh; safe: stream order)
  float* o0   = ws + (size_t)2 * B_SZ * N3D;         // B*64
  float* comb = o0 + (size_t)B_SZ * D_SZ;            // NI
  int*   topi = (int*)(comb + NI_SZ);                // 5

  // 8 waves/block, one 16x16 tile per wave
  k_gemm_gru<<<(B_SZ / 16) * (N3D / 16) / 8, 256, 0, stream>>>(
      u, v, delta_t, edge, uemb, iemb, mem, w_ih, w_hh, b_ih, b_hh, gi, gh);
  k_gru_elem<<<(B_SZ * D_SZ) / 256, 256, 0, stream>>>(gi, gh, u, mem, h_new);

  k_qkv<<<(2 * B_SZ / 16) * (N3D / 16) / 8, 256, 0, stream>>>(
      u, v, edge, uemb, iemb, w_in, b_in, qkv);
  k_attn<<<B_SZ, 64, 0, stream>>>(qkv, o0);

  k_outproj<<<(B_SZ / 16) * (D_SZ / 16) / 8, 256, 0, stream>>>(o0, w_out, b_out,
                                                               z_u);

  k_cov<<<NI_SZ, 256, 0, stream>>>(cov, v, returns, comb);
  k_top5<<<1, 256, 0, stream>>>(comb, topi);
  k_select<<<(B_SZ + 255) / 256, 256, 0, stream>>>(topi, v, pos, neg);
}